// SparsemaxAlignmentLoss_59742995087647
// MI455X (gfx1250) — compile-verified
//
#include <hip/hip_runtime.h>
#include <hip/hip_bf16.h>

// Problem constants (from reference: B=64, K=16, L=16384)
#define L_DIM   16384
#define BATCH   64
#define HEADS   16
#define NROWS   (BATCH * HEADS)   // 1024
#define TPB     256               // 8 wave32 per block
#define VPT     16                // float4 chunks per thread (16*4 = 64 elems)
#define PER     64                // f32 elements per thread
#define NWAVES  (TPB / 32)        // 8
#define BISECT_ITERS 26           // interval 1.0 -> ~1.5e-8, below f32 ulp of data

typedef __attribute__((ext_vector_type(2))) float v2f;
typedef __attribute__((ext_vector_type(8))) float v8f;

// ---------------- wave32 reductions ----------------
__device__ __forceinline__ float wave_sum(float v) {
#pragma unroll
  for (int o = 16; o > 0; o >>= 1) v += __shfl_xor(v, o, 32);
  return v;
}
__device__ __forceinline__ float wave_max(float v) {
#pragma unroll
  for (int o = 16; o > 0; o >>= 1) v = fmaxf(v, __shfl_xor(v, o, 32));
  return v;
}

// Full-wave (32 lane) sum via V_WMMA_F32_16X16X4_F32.
// A: lane m (0..15) -> A[m][0]=v, A[m][1]=0 ; lane m+16 -> A[m][2]=v, A[m][3]=0.
// B: all ones. => D[m][n] = v[m] + v[m+16] for every n.
// Lane l holds D rows r (+8 for upper half) in 8 accum VGPRs; summing them and
// adding across the xor-16 pair yields the full 32-lane sum on every lane.
// EXEC must be all-ones at the call site (uniform control flow only).
__device__ __forceinline__ float wmma_wave_sum(float v) {
  v2f a; a.x = v;    a.y = 0.0f;
  v2f b; b.x = 1.0f; b.y = 1.0f;
  v8f c = {};
  c = __builtin_amdgcn_wmma_f32_16x16x4_f32(false, a, false, b,
                                            (short)0, c, false, false);
  float s = c[0] + c[1] + c[2] + c[3] + c[4] + c[5] + c[6] + c[7];
  s += __shfl_xor(s, 16, 32);
  return s;
}

__device__ __forceinline__ float block_sum(float v, float* s_red, int t) {
  v = wave_sum(v);
  __syncthreads();                      // protect s_red reuse
  if ((t & 31) == 0) s_red[t >> 5] = v;
  __syncthreads();
  float r = 0.0f;
#pragma unroll
  for (int w = 0; w < NWAVES; ++w) r += s_red[w];  // identical order on all threads
  return r;
}
__device__ __forceinline__ float block_max(float v, float* s_red, int t) {
  v = wave_max(v);
  __syncthreads();
  if ((t & 31) == 0) s_red[t >> 5] = v;
  __syncthreads();
  float r = -INFINITY;
#pragma unroll
  for (int w = 0; w < NWAVES; ++w) r = fmaxf(r, s_red[w]);
  return r;
}

// ---------------- kernel A: per-batch mask sums ----------------
__global__ void mask_sum_kernel(const float* __restrict__ mask,
                                float* __restrict__ qsum) {
  __shared__ float s_red[NWAVES];
  const int t = threadIdx.x;
  const int b = blockIdx.x;
  const float4* m4 = (const float4*)(mask + (size_t)b * L_DIM);
  float acc = 0.0f;
#pragma unroll
  for (int i = 0; i < VPT; ++i) {
    float4 v = m4[t + i * TPB];
    acc += (v.x + v.y) + (v.z + v.w);
  }
  acc = block_sum(acc, s_red, t);
  if (t == 0) qsum[b] = acc;
}

// ---------------- kernel B: one block per (b,k) row ----------------
// LDS layout: [0, L_DIM) staged mask row (async copy), [L_DIM, L_DIM+16) reduce scratch
__global__ void __launch_bounds__(TPB)
sparsemax_row_kernel(const float* __restrict__ logits,
                     const float* __restrict__ mask,
                     const float* __restrict__ qsum,
                     float* __restrict__ row_loss) {
  extern __shared__ __align__(16) float smem[];
  float* s_red = smem + L_DIM;

  const int t   = threadIdx.x;
  const int row = blockIdx.x;          // 0..1023
  const int b   = row >> 4;            // row / HEADS
  const float* zrow = logits + (size_t)row * L_DIM;
  const float* mrow = mask   + (size_t)b   * L_DIM;

  // --- kick off async mask-row staging into LDS (hidden behind bisection) ---
  {
    const unsigned lds_base = (unsigned)(size_t)(void*)smem;  // low 32b = LDS offset
#pragma unroll
    for (int i = 0; i < VPT; ++i) {
      const unsigned byte_off = (unsigned)(t + i * TPB) * 16u;
      const unsigned lds_off  = lds_base + byte_off;
      const void*    gaddr    = (const void*)((const char*)mrow + byte_off);
      asm volatile("global_load_async_to_lds_b128 %0, %1, off"
                   :: "v"(lds_off), "v"(gaddr) : "memory");
    }
  }

  // --- load row into registers (read global exactly once) ---
  float z[PER];
  {
    const float4* z4 = (const float4*)zrow;
#pragma unroll
    for (int i = 0; i < VPT; ++i) {
      float4 v = z4[t + i * TPB];
      z[4 * i + 0] = v.x; z[4 * i + 1] = v.y;
      z[4 * i + 2] = v.z; z[4 * i + 3] = v.w;
    }
  }

  // --- row max ---
  float m = -INFINITY;
#pragma unroll
  for (int i = 0; i < PER; ++i) m = fmaxf(m, z[i]);
  const float zmax = block_max(m, s_red, t);

  // --- bisection on g(tau) = sum relu(z - tau) - 1, tau in [zmax-1, zmax] ---
  float lo = zmax - 1.0f, hi = zmax;
#pragma unroll 1
  for (int it = 0; it < BISECT_ITERS; ++it) {
    const float mid = 0.5f * (lo + hi);
    float acc = 0.0f;
#pragma unroll
    for (int i = 0; i < PER; ++i) acc += fmaxf(z[i] - mid, 0.0f);
    const float g = block_sum(acc, s_red, t) - 1.0f;  // identical on all threads
    if (g > 0.0f) lo = mid; else hi = mid;
  }

  // --- exact tau from the (now exact) support set ---
  const float mid = 0.5f * (lo + hi);
  float cnt = 0.0f, ssum = 0.0f;
#pragma unroll
  for (int i = 0; i < PER; ++i) {
    const float gt = (z[i] > mid) ? 1.0f : 0.0f;
    cnt  += gt;
    ssum  = __builtin_fmaf(gt, z[i], ssum);
  }
  const float C = block_sum(cnt,  s_red, t);
  const float S = block_sum(ssum, s_red, t);
  const float tau = (S - 1.0f) / fmaxf(C, 1.0f);

  // --- wait for the async-staged mask row, then fused loss pass ---
  asm volatile("s_wait_asynccnt 0" ::: "memory");
  __syncthreads();

  const float qs   = qsum[b];
  const float qinv = 1.0f / fmaxf(qs, 1e-12f);
  const float4* q4 = (const float4*)smem;
  float acc = 0.0f;
#pragma unroll
  for (int i = 0; i < VPT; ++i) {
    const float4 mv = q4[t + i * TPB];
    const float d0 = fmaxf(z[4 * i + 0] - tau, 0.0f) - mv.x * qinv;
    const float d1 = fmaxf(z[4 * i + 1] - tau, 0.0f) - mv.y * qinv;
    const float d2 = fmaxf(z[4 * i + 2] - tau, 0.0f) - mv.z * qinv;
    const float d3 = fmaxf(z[4 * i + 3] - tau, 0.0f) - mv.w * qinv;
    acc += d0 * d0 + d1 * d1 + d2 * d2 + d3 * d3;
  }

  // --- block reduce via WMMA wave sums ---
  const float wsum = wmma_wave_sum(acc);     // uniform flow: EXEC all ones
  __syncthreads();
  if ((t & 31) == 0) s_red[t >> 5] = wsum;
  __syncthreads();
  if (t == 0) {
    float tot = 0.0f;
#pragma unroll
    for (int w = 0; w < NWAVES; ++w) tot += s_red[w];
    row_loss[row] = tot;
  }
}

// ---------------- kernel C: deterministic final reduction ----------------
__global__ void final_reduce_kernel(const float* __restrict__ row_loss,
                                    float* __restrict__ out) {
  __shared__ float s_red[NWAVES];
  const int t = threadIdx.x;
  float v = row_loss[t] + row_loss[t + 256] + row_loss[t + 512] + row_loss[t + 768];
  const float wsum = wmma_wave_sum(v);
  if ((t & 31) == 0) s_red[t >> 5] = wsum;
  __syncthreads();
  if (t == 0) {
    float tot = 0.0f;
#pragma unroll
    for (int w = 0; w < NWAVES; ++w) tot += s_red[w];
    out[0] = 0.5f * tot * (1.0f / (float)NROWS);   // 0.5 * mean over (B,K)
  }
}

extern "C" void kernel_launch(void* const* d_in, const int* in_sizes, int n_in,
                              void* d_out, int out_size, void* d_ws, size_t ws_size,
                              hipStream_t stream) {
  const float* logits = (const float*)d_in[0];   // (B, K, L) f32
  const float* mask   = (const float*)d_in[1];   // (B, L)    f32
  float* out  = (float*)d_out;                   // scalar f32
  float* ws   = (float*)d_ws;
  float* qsum = ws;            // 64 floats
  float* rlos = ws + BATCH;    // 1024 floats

  mask_sum_kernel<<<BATCH, TPB, 0, stream>>>(mask, qsum);

  const size_t shmem = (size_t)(L_DIM + 16) * sizeof(float); // 64KB stage + scratch
  sparsemax_row_kernel<<<NROWS, TPB, shmem, stream>>>(logits, mask, qsum, rlos);

  final_reduce_kernel<<<1, TPB, 0, stream>>>(rlos, out);
}